// GraphSAGEModel_27058293965203
// MI455X (gfx1250) — compile-verified
//
#include <hip/hip_runtime.h>

#define N_NODES 50000
#define N_EDGES 800000
#define N_PRED  200000
#define DIM     128
#define PSTR    288   // dwords per K-pair group in LDS: 256 data + 32 pad
                      // (288 % 64 == 32 -> the two half-waves of a wave32 hit
                      //  disjoint 32-bank halves on ds_load_b64: conflict-free)

typedef __attribute__((ext_vector_type(2))) float v2f;
typedef __attribute__((ext_vector_type(8))) float v8f;

// ---------------------------------------------------------------- zero fill
__global__ void zero_f32_kernel(float* __restrict__ p, int n) {
    int i = blockIdx.x * blockDim.x + threadIdx.x;
    if (i < n) p[i] = 0.0f;
}

// ---------------------------------------------------------------- degree
__global__ void degree_kernel(const int* __restrict__ dst,
                              float* __restrict__ deg, int e_count) {
    int e = blockIdx.x * blockDim.x + threadIdx.x;
    if (e < e_count) atomicAdd(&deg[dst[e]], 1.0f);
}

// ---------------------------------------------------------------- gather + scatter-add
// one block (128 threads) per edge; src/dst loads are uniform -> scalar loads
__global__ __launch_bounds__(DIM) void scatter_kernel(
    const float* __restrict__ feat,
    const int* __restrict__ src,
    const int* __restrict__ dst,
    float* __restrict__ agg) {
    int e = blockIdx.x;
    int c = threadIdx.x;
    int s = src[e];
    int d = dst[e];
    atomicAdd(&agg[(long)d * DIM + c], feat[(long)s * DIM + c]);
}

// ---------------------------------------------------------------- fused SAGE layer GEMM
// out[r,:] = relu?( (agg[r,:]/max(deg,1)) @ Wl + x[r,:] @ Wr + b )
// 256 threads = 8 waves; each wave computes a 16x128 slab via V_WMMA_F32_16X16X4_F32.
// Weights live in LDS in K-pair-interleaved layout: sW[k/2][col][2] so each
// B fragment is a single aligned ds_load_b64 feeding WMMA SRC1 directly.
__global__ __launch_bounds__(256) void sage_gemm_kernel(
    const float* __restrict__ agg,   // N x 128 (un-normalized sums)
    const float* __restrict__ deg,   // N
    const float* __restrict__ xin,   // N x 128 (self features)
    const float* __restrict__ Wl,    // 128 x 128 row-major [k][n]
    const float* __restrict__ Wr,    // 128 x 128 row-major [k][n]
    const float* __restrict__ bias,  // 128
    float* __restrict__ out,         // N x 128
    int n_nodes, int do_relu)
{
    __shared__ float sWl[(DIM / 2) * PSTR];   // 64 K-pair groups
    __shared__ float sWr[(DIM / 2) * PSTR];

    // stage weights: sW[p*PSTR + 2*col + j] = W[(2p+j)*128 + col]
    for (int idx = threadIdx.x; idx < (DIM / 2) * DIM; idx += 256) {
        int p   = idx >> 7;       // K-pair group 0..63
        int col = idx & 127;      // consecutive threads -> consecutive cols (coalesced)
        float2 a, b;
        a.x = Wl[(2 * p) * DIM + col];
        a.y = Wl[(2 * p + 1) * DIM + col];
        b.x = Wr[(2 * p) * DIM + col];
        b.y = Wr[(2 * p + 1) * DIM + col];
        *(float2*)&sWl[p * PSTR + 2 * col] = a;
        *(float2*)&sWr[p * PSTR + 2 * col] = b;
    }
    __syncthreads();

    const int wave = threadIdx.x >> 5;
    const int lane = threadIdx.x & 31;
    const int half = lane >> 4;          // 0: K pair {0,1} ; 1: K pair {2,3}
    const int l16  = lane & 15;

    const int rowBase = blockIdx.x * 128 + wave * 16;
    int aRow  = rowBase + l16;                         // this lane's A row (M)
    int aRowC = aRow < n_nodes ? aRow : (n_nodes - 1); // clamp: keeps EXEC all-1s
    const float inv = 1.0f / fmaxf(deg[aRowC], 1.0f);

    const float* __restrict__ aggRow = agg + (long)aRowC * DIM;
    const float* __restrict__ xRow   = xin + (long)aRowC * DIM;

    v8f acc[8];
#pragma unroll
    for (int nt = 0; nt < 8; ++nt) { v8f zz = {}; acc[nt] = zz; }

#pragma unroll 4
    for (int kk = 0; kk < DIM / 4; ++kk) {
        const int k0 = kk * 4 + half * 2;
        const int p  = kk * 2 + half;          // this half-wave's K-pair group
        // A fragments: agg (mean-fused) and self features (global_load_b64 each)
        float2 ga = *(const float2*)&aggRow[k0];
        float2 gx = *(const float2*)&xRow[k0];
        v2f aAgg; aAgg.x = ga.x * inv; aAgg.y = ga.y * inv;
        v2f aX;   aX.x   = gx.x;      aX.y   = gx.y;
        const float* __restrict__ blp = &sWl[p * PSTR];
        const float* __restrict__ brp = &sWr[p * PSTR];
#pragma unroll
        for (int nt = 0; nt < 8; ++nt) {
            const int col2 = (nt * 16 + l16) * 2;
            v2f bl = *(const v2f*)&blp[col2];   // one ds_load_b64, conflict-free
            v2f br = *(const v2f*)&brp[col2];
            acc[nt] = __builtin_amdgcn_wmma_f32_16x16x4_f32(
                false, aAgg, false, bl, (short)0, acc[nt], false, false);
            acc[nt] = __builtin_amdgcn_wmma_f32_16x16x4_f32(
                false, aX,   false, br, (short)0, acc[nt], false, false);
        }
    }

    // D layout: vgpr i -> row rowBase + i + 8*half, col = nt*16 + l16
#pragma unroll
    for (int nt = 0; nt < 8; ++nt) {
        const int col = nt * 16 + l16;
        const float bv = bias[col];
#pragma unroll
        for (int i = 0; i < 8; ++i) {
            int r = rowBase + i + half * 8;
            if (r < n_nodes) {
                float v = acc[nt][i] + bv;
                if (do_relu) v = fmaxf(v, 0.0f);
                out[(long)r * DIM + col] = v;
            }
        }
    }
}

// ---------------------------------------------------------------- pred-edge dot
// one wave per edge: 32 lanes x float4 = 128 channels, shuffle reduce
__global__ __launch_bounds__(256) void edge_dot_kernel(
    const float* __restrict__ z,
    const int* __restrict__ ps,
    const int* __restrict__ pd,
    float* __restrict__ out, int ep) {
    int e    = blockIdx.x * 8 + (threadIdx.x >> 5);
    int lane = threadIdx.x & 31;
    if (e >= ep) return;
    int a = ps[e], b = pd[e];
    float4 va = *(const float4*)(z + (long)a * DIM + lane * 4);
    float4 vb = *(const float4*)(z + (long)b * DIM + lane * 4);
    float s = va.x * vb.x + va.y * vb.y + va.z * vb.z + va.w * vb.w;
#pragma unroll
    for (int off = 16; off > 0; off >>= 1) s += __shfl_down(s, off, 32);
    if (lane == 0) out[e] = s;
}

// ---------------------------------------------------------------- host launcher
extern "C" void kernel_launch(void* const* d_in, const int* in_sizes, int n_in,
                              void* d_out, int out_size, void* d_ws, size_t ws_size,
                              hipStream_t stream) {
    const float* x   = (const float*)d_in[0];
    const float* W1l = (const float*)d_in[1];
    const float* b1  = (const float*)d_in[2];
    const float* W1r = (const float*)d_in[3];
    const float* W2l = (const float*)d_in[4];
    const float* b2  = (const float*)d_in[5];
    const float* W2r = (const float*)d_in[6];
    const int*   ei  = (const int*)d_in[7];   // (2, E) int32
    const int*   pe  = (const int*)d_in[8];   // (2, EP) int32
    float* logits    = (float*)d_out;

    const int* src = ei;
    const int* dst = ei + N_EDGES;
    const int* ps  = pe;
    const int* pd  = pe + N_PRED;

    // workspace layout: deg | agg | h | z  (~77 MB)
    float* deg = (float*)d_ws;
    float* agg = deg + N_NODES;
    float* h   = agg + (size_t)N_NODES * DIM;
    float* z   = h   + (size_t)N_NODES * DIM;

    const int zeroAll = N_NODES + N_NODES * DIM;   // deg + agg contiguous
    const int gemmBlocks = (N_NODES + 127) / 128;  // 391

    // ---- layer 1
    zero_f32_kernel<<<(zeroAll + 255) / 256, 256, 0, stream>>>(deg, zeroAll);
    degree_kernel<<<(N_EDGES + 255) / 256, 256, 0, stream>>>(dst, deg, N_EDGES);
    scatter_kernel<<<N_EDGES, DIM, 0, stream>>>(x, src, dst, agg);
    sage_gemm_kernel<<<gemmBlocks, 256, 0, stream>>>(agg, deg, x, W1l, W1r, b1,
                                                     h, N_NODES, /*relu=*/1);
    // ---- layer 2 (degree unchanged; re-zero agg only)
    zero_f32_kernel<<<(N_NODES * DIM + 255) / 256, 256, 0, stream>>>(agg, N_NODES * DIM);
    scatter_kernel<<<N_EDGES, DIM, 0, stream>>>(h, src, dst, agg);
    sage_gemm_kernel<<<gemmBlocks, 256, 0, stream>>>(agg, deg, h, W2l, W2r, b2,
                                                     z, N_NODES, /*relu=*/0);
    // ---- link prediction logits
    edge_dot_kernel<<<(N_PRED + 7) / 8, 256, 0, stream>>>(z, ps, pd, logits, N_PRED);
}